// OtherRNN_24541443129810
// MI455X (gfx1250) — compile-verified
//
#include <hip/hip_runtime.h>

typedef __attribute__((ext_vector_type(2))) float v2f;
typedef __attribute__((ext_vector_type(4))) float v4f;
typedef __attribute__((ext_vector_type(8))) float v8f;

static constexpr int Bsz  = 512;
static constexpr int Lsz  = 4096;
static constexpr int OBS  = 13;
static constexpr int ACT  = 3;
static constexpr int Hdim = 10;
static constexpr int IN   = OBS + ACT;   // 16
static constexpr int ZDIM = IN + Hdim;   // 26
static constexpr int TILE = 16;

__device__ __forceinline__ v8f wmma_f32(v2f a, v2f b, v8f c) {
  // D = A(16x4 f32) * B(4x16 f32) + C(16x16 f32)
  return __builtin_amdgcn_wmma_f32_16x16x4_f32(
      /*neg_a=*/false, a, /*neg_b=*/false, b,
      /*c_mod=*/(short)0, c, /*reuse_a=*/false, /*reuse_b=*/false);
}

__global__ __launch_bounds__(32)
void rnn_scan_kernel(const float* __restrict__ x,    // [B, L, 16]
                     const float* __restrict__ dt,   // [B, L]
                     const float* __restrict__ Wgh,  // [13,10]
                     const float* __restrict__ bgh,  // [10]
                     const float* __restrict__ Wrec, // [26,10]
                     const float* __restrict__ brec, // [10]
                     const float* __restrict__ W1,   // [10,10]
                     const float* __restrict__ b1,   // [10]
                     const float* __restrict__ W2,   // [10,13]
                     const float* __restrict__ b2,   // [13]
                     float* __restrict__ out)        // [B, L, 13]
{
  __shared__ __align__(16) float ldsH[TILE * 16];  // h  as [M=16][N=16]
  __shared__ __align__(16) float ldsU[TILE * 16];  // u  as [M=16][N=16]
  __shared__ __align__(16) float ldsDT[32];        // dt; slots 16..31 unused

  const int lane = threadIdx.x & 31;
  const int n    = lane & 15;   // N-column in C/D layout; batch-row for A loads
  const int half = lane >> 4;   // 0: lanes 0-15, 1: lanes 16-31
  const long bBase = (long)blockIdx.x * TILE;

  // ---- loop-invariant B fragments (K x N row-major weights, zero padded) ----
  // B layout (f32 4x16): VGPR0 = row k0 (lanes 0-15) / row k0+2 (lanes 16-31),
  //                      VGPR1 = row k0+1            / row k0+3
  auto loadB = [&](const float* W, int K, int N, int k0) -> v2f {
    v2f r; r.x = 0.f; r.y = 0.f;
    const int kA = k0 + 2 * half;
    if (n < N) {
      if (kA     < K) r.x = W[(long)kA * N + n];
      if (kA + 1 < K) r.y = W[(long)(kA + 1) * N + n];
    }
    return r;
  };

  v2f Bgh[4], Brec[7], B1f[3], B2f[3];
#pragma unroll
  for (int j = 0; j < 4; ++j) Bgh[j]  = loadB(Wgh,  OBS,  Hdim, 4 * j);
#pragma unroll
  for (int j = 0; j < 7; ++j) Brec[j] = loadB(Wrec, ZDIM, Hdim, 4 * j);
#pragma unroll
  for (int j = 0; j < 3; ++j) B1f[j]  = loadB(W1,   Hdim, Hdim, 4 * j);
#pragma unroll
  for (int j = 0; j < 3; ++j) B2f[j]  = loadB(W2,   Hdim, OBS,  4 * j);

  const float bghL  = (n < Hdim) ? bgh[n]  : 0.f;
  const float brecL = (n < Hdim) ? brec[n] : 0.f;
  const float b1L   = (n < Hdim) ? b1[n]   : 0.f;
  const float b2L   = (n < OBS)  ? b2[n]   : 0.f;

  // per-lane streaming pointers (this lane owns tile row `n` for A loads)
  const float* xrow   = x  + (bBase + n) * (long)Lsz * IN;
  const float* dtrow  = dt + (bBase + n) * (long)Lsz;
  float* outBase      = out + ((bBase + 8 * half) * (long)Lsz) * OBS + n;

  // ---- h0 = x[:,0,:13] @ Wgh + bgh  (act cols 13..15 killed by zero B rows)
  v8f hC;
#pragma unroll
  for (int v = 0; v < 8; ++v) hC[v] = bghL;
  {
    v2f xa0[4];
#pragma unroll
    for (int j = 0; j < 4; ++j)
      xa0[j] = *(const v2f*)(xrow + 4 * j + 2 * half);
#pragma unroll
    for (int j = 0; j < 4; ++j) hC = wmma_f32(xa0[j], Bgh[j], hC);
  }

  // publish h0 and pull its A-layout fragments
#pragma unroll
  for (int v = 0; v < 8; ++v) ldsH[(v + 8 * half) * 16 + n] = hC[v];
  __syncthreads();
  v2f hA[3];
#pragma unroll
  for (int j = 0; j < 3; ++j)
    hA[j] = *(const v2f*)(ldsH + n * 16 + 4 * j + 2 * half);

  for (int t = 0; t < Lsz; ++t) {
    const float* xt = xrow + (long)t * IN;
    // prefetch x a few steps ahead; keep in near caches (WGP scope)
    __builtin_prefetch(xt + 8 * IN, 0, 3);

    // ---- A fragments of z = [x_t | h_t]  (K = 28, rows 26..27 zero in B)
    v2f xa[4];
#pragma unroll
    for (int j = 0; j < 4; ++j)
      xa[j] = *(const v2f*)(xt + 4 * j + 2 * half);

    // all lanes store dt (lanes 16..31 hit unread slots -> no divergence)
    ldsDT[lane] = dtrow[t];
    __syncthreads();

    v8f acc;
#pragma unroll
    for (int v = 0; v < 8; ++v) acc[v] = 0.f;
#pragma unroll
    for (int j = 0; j < 4; ++j) acc = wmma_f32(xa[j], Brec[j], acc);
#pragma unroll
    for (int j = 0; j < 3; ++j) acc = wmma_f32(hA[j], Brec[4 + j], acc);

    // ---- Euler update: h[v] += dt(row v+8*half) * (acc[v] + brec)
    const v4f dA = *(const v4f*)(ldsDT + 8 * half);
    const v4f dB = *(const v4f*)(ldsDT + 8 * half + 4);
    hC[0] += dA.x * (acc[0] + brecL);
    hC[1] += dA.y * (acc[1] + brecL);
    hC[2] += dA.z * (acc[2] + brecL);
    hC[3] += dA.w * (acc[3] + brecL);
    hC[4] += dB.x * (acc[4] + brecL);
    hC[5] += dB.y * (acc[5] + brecL);
    hC[6] += dB.z * (acc[6] + brecL);
    hC[7] += dB.w * (acc[7] + brecL);

    // ---- publish h_{t+1}; its A-fragments feed h@W1 now and z at t+1
    __syncthreads();
#pragma unroll
    for (int v = 0; v < 8; ++v) ldsH[(v + 8 * half) * 16 + n] = hC[v];
    __syncthreads();
#pragma unroll
    for (int j = 0; j < 3; ++j)
      hA[j] = *(const v2f*)(ldsH + n * 16 + 4 * j + 2 * half);

    // ---- u = relu(h @ W1 + b1)
    v8f u;
#pragma unroll
    for (int v = 0; v < 8; ++v) u[v] = b1L;
#pragma unroll
    for (int j = 0; j < 3; ++j) u = wmma_f32(hA[j], B1f[j], u);
#pragma unroll
    for (int v = 0; v < 8; ++v) u[v] = fmaxf(u[v], 0.f);

    // ---- transpose u through LDS, then o = u @ W2 + b2
#pragma unroll
    for (int v = 0; v < 8; ++v) ldsU[(v + 8 * half) * 16 + n] = u[v];
    __syncthreads();
    v2f ua[3];
#pragma unroll
    for (int j = 0; j < 3; ++j)
      ua[j] = *(const v2f*)(ldsU + n * 16 + 4 * j + 2 * half);

    v8f o;
#pragma unroll
    for (int v = 0; v < 8; ++v) o[v] = b2L;
#pragma unroll
    for (int j = 0; j < 3; ++j) o = wmma_f32(ua[j], B2f[j], o);

    // ---- store out[b, t, 0:13]; lane column n, rows v + 8*half
    if (n < OBS) {
      float* op = outBase + (long)t * OBS;
#pragma unroll
      for (int v = 0; v < 8; ++v)
        op[(long)v * Lsz * OBS] = o[v];
    }
    __syncthreads();
  }
}

extern "C" void kernel_launch(void* const* d_in, const int* in_sizes, int n_in,
                              void* d_out, int out_size, void* d_ws, size_t ws_size,
                              hipStream_t stream) {
  const float* x    = (const float*)d_in[0];
  const float* dt   = (const float*)d_in[1];
  const float* Wgh  = (const float*)d_in[2];
  const float* bgh  = (const float*)d_in[3];
  const float* Wrec = (const float*)d_in[4];
  const float* brec = (const float*)d_in[5];
  const float* W1   = (const float*)d_in[6];
  const float* b1   = (const float*)d_in[7];
  const float* W2   = (const float*)d_in[8];
  const float* b2   = (const float*)d_in[9];
  float* out = (float*)d_out;

  dim3 grid(Bsz / TILE);   // 32 tiles of 16 batch rows
  dim3 block(32);          // one wave32 per tile
  rnn_scan_kernel<<<grid, block, 0, stream>>>(x, dt, Wgh, bgh, Wrec, brec,
                                              W1, b1, W2, b2, out);
}